// GraphNet_59158879535366
// MI455X (gfx1250) — compile-verified
//
#include <hip/hip_runtime.h>

typedef float v2f __attribute__((ext_vector_type(2)));
typedef float v8f __attribute__((ext_vector_type(8)));

#define NFEAT 128
#define HID   16
#define MLPH  100
#define NCLS  12
#define NGR   512

// ---------------------------------------------------------------------------
// WMMA GEMM: Y[M,16] = X[M,K] @ W[K,16], fp32 via V_WMMA_F32_16X16X4_F32.
// One wave32 computes one 16x16 output tile (16 node rows x 16 hidden cols).
// A-layout (16x4 f32): lanes 0-15: M=lane, K={k,k+1}; lanes 16-31: M=lane-16, K={k+2,k+3}
// B-layout (4x16 f32): V0/V1: lanes 0-15 hold rows K={k,k+1}, lanes 16-31 rows K={k+2,k+3}
// D-layout (16x16 f32): VGPR v: lanes 0-15 -> row v, lanes 16-31 -> row v+8
// ---------------------------------------------------------------------------
template <int K>
__global__ void gemm16_wmma_kernel(const float* __restrict__ X,
                                   const float* __restrict__ W,
                                   float* __restrict__ Y, int tiles) {
  int wave = (int)((blockIdx.x * blockDim.x + threadIdx.x) >> 5);  // wave-uniform
  if (wave >= tiles) return;                                       // uniform exit: EXEC stays all-1s
  int lane = threadIdx.x & 31;
  int half = lane >> 4;
  int l    = lane & 15;
  int m0   = wave << 4;

  const float* xrow = X + (size_t)(m0 + l) * K;
  v8f acc = {0.f, 0.f, 0.f, 0.f, 0.f, 0.f, 0.f, 0.f};

#pragma unroll
  for (int k = 0; k < K; k += 4) {
    int ka = k + 2 * half;
    v2f a, b;
    a.x = xrow[ka];
    a.y = xrow[ka + 1];
    b.x = W[ka * HID + l];
    b.y = W[(ka + 1) * HID + l];
    acc = __builtin_amdgcn_wmma_f32_16x16x4_f32(false, a, false, b,
                                                (short)0, acc, false, false);
  }
#pragma unroll
  for (int v = 0; v < 8; ++v)
    Y[(size_t)(m0 + v + 8 * half) * HID + l] = acc[v];
}

// Tail rows (M % 16) fallback — unused for M=100000 but kept for generality.
template <int K>
__global__ void gemm16_tail_kernel(const float* __restrict__ X,
                                   const float* __restrict__ W,
                                   float* __restrict__ Y, int m_start, int M) {
  int idx = blockIdx.x * blockDim.x + threadIdx.x;
  int row = m_start + (idx >> 4);
  int col = idx & 15;
  if (row >= M) return;
  float s = 0.f;
  for (int k = 0; k < K; ++k) s += X[(size_t)row * K + k] * W[k * HID + col];
  Y[(size_t)row * HID + col] = s;
}

// ---------------------------------------------------------------------------
// Degree / normalization
// ---------------------------------------------------------------------------
__global__ void deg_init_kernel(float* deg, int n) {
  int i = blockIdx.x * blockDim.x + threadIdx.x;
  if (i < n) deg[i] = 1.0f;  // self-loop contribution
}

__global__ void deg_accum_kernel(const int* __restrict__ ei, float* deg, int E) {
  int e = blockIdx.x * blockDim.x + threadIdx.x;
  if (e < E) atomicAdd(&deg[ei[E + e]], 1.0f);  // dst row of edge_index
}

__global__ void dinv_kernel(float* d, int n) {
  int i = blockIdx.x * blockDim.x + threadIdx.x;
  if (i < n) {
    float v = d[i];
    d[i] = (v > 0.f) ? rsqrtf(v) : 0.f;
  }
}

// ---------------------------------------------------------------------------
// GCN aggregation: out = segment_sum(h[src]*norm, dst) incl. self loops
// ---------------------------------------------------------------------------
__global__ void self_loop_init_kernel(const float* __restrict__ hpre,
                                      const float* __restrict__ dinv,
                                      float* __restrict__ out, int n) {
  int idx = blockIdx.x * blockDim.x + threadIdx.x;
  if (idx < n * HID) {
    int i = idx >> 4;
    float di = dinv[i];
    out[idx] = hpre[idx] * di * di;  // self-loop: norm = dinv[i]^2
  }
}

__global__ void edge_scatter_kernel(const int* __restrict__ ei,
                                    const float* __restrict__ hpre,
                                    const float* __restrict__ dinv,
                                    float* __restrict__ out, int E) {
  int idx = blockIdx.x * blockDim.x + threadIdx.x;
  if (idx >= E * HID) return;
  int e = idx >> 4;
  int f = idx & 15;
  int s = ei[e];
  int d = ei[E + e];
  float nrm = dinv[s] * dinv[d];
  atomicAdd(&out[d * HID + f], hpre[s * HID + f] * nrm);
}

__global__ void bias_relu_kernel(float* __restrict__ out,
                                 const float* __restrict__ b, int n) {
  int idx = blockIdx.x * blockDim.x + threadIdx.x;
  if (idx < n * HID) {
    float v = out[idx] + b[idx & 15];
    out[idx] = v > 0.f ? v : 0.f;
  }
}

// ---------------------------------------------------------------------------
// Pool + MLP head
// ---------------------------------------------------------------------------
__global__ void pool_zero_kernel(float* pooled) {
  int i = blockIdx.x * blockDim.x + threadIdx.x;
  if (i < NGR * HID) pooled[i] = 0.f;
}

__global__ void pool_kernel(const int* __restrict__ batch,
                            const float* __restrict__ h,
                            float* __restrict__ pooled, int n) {
  int idx = blockIdx.x * blockDim.x + threadIdx.x;
  if (idx < n * HID) {
    int i = idx >> 4;
    int f = idx & 15;
    atomicAdd(&pooled[batch[i] * HID + f], h[idx]);
  }
}

__global__ void mlp1_kernel(const float* __restrict__ pooled,
                            const float* __restrict__ lw1,
                            const float* __restrict__ lb1,
                            float* __restrict__ hid) {
  int idx = blockIdx.x * blockDim.x + threadIdx.x;
  if (idx >= NGR * MLPH) return;
  int g = idx / MLPH;
  int j = idx % MLPH;
  float s = lb1[j];
  const float* p = pooled + g * HID;
#pragma unroll
  for (int f = 0; f < HID; ++f) {
    float pv = p[f];
    pv = pv > 0.f ? pv : 0.f;  // g = relu(pooled)
    s += pv * lw1[f * MLPH + j];
  }
  hid[idx] = s > 0.f ? s : 0.f;  // relu
}

__global__ void mlp2_kernel(const float* __restrict__ hid,
                            const float* __restrict__ lw2,
                            const float* __restrict__ lb2,
                            float* __restrict__ out) {
  int idx = blockIdx.x * blockDim.x + threadIdx.x;
  if (idx >= NGR * NCLS) return;
  int g = idx / NCLS;
  int c = idx % NCLS;
  float s = lb2[c];
  const float* h = hid + g * MLPH;
  for (int j = 0; j < MLPH; ++j) s += h[j] * lw2[j * NCLS + c];
  out[idx] = s;
}

// ---------------------------------------------------------------------------
static inline int cdiv(int a, int b) { return (a + b - 1) / b; }

extern "C" void kernel_launch(void* const* d_in, const int* in_sizes, int n_in,
                              void* d_out, int out_size, void* d_ws, size_t ws_size,
                              hipStream_t stream) {
  const float* x      = (const float*)d_in[0];
  const int*   ei     = (const int*)d_in[1];   // [2, E]
  const int*   batch  = (const int*)d_in[2];   // [N]
  const float* W1     = (const float*)d_in[3];
  const float* b1     = (const float*)d_in[4];
  const float* W2     = (const float*)d_in[5];
  const float* b2     = (const float*)d_in[6];
  const float* lw1    = (const float*)d_in[7];
  const float* lb1    = (const float*)d_in[8];
  const float* lw2    = (const float*)d_in[9];
  const float* lb2    = (const float*)d_in[10];
  float*       out    = (float*)d_out;

  const int N = in_sizes[0] / NFEAT;   // 100000
  const int E = in_sizes[1] / 2;       // 3200000

  // Workspace layout (floats): dinv[N] | A[N*16] | B[N*16] | pooled | hid
  float* dinv   = (float*)d_ws;
  float* A      = dinv + N;            // h_pre (GEMM output)
  float* B      = A + (size_t)N * HID; // aggregation / activated features
  float* pooled = B + (size_t)N * HID;
  float* hid    = pooled + NGR * HID;

  const int T = 256;
  const int tiles = N / 16;
  const int tail  = N - tiles * 16;

  // 1) degrees -> dinv (in place)
  deg_init_kernel<<<cdiv(N, T), T, 0, stream>>>(dinv, N);
  deg_accum_kernel<<<cdiv(E, T), T, 0, stream>>>(ei, dinv, E);
  dinv_kernel<<<cdiv(N, T), T, 0, stream>>>(dinv, N);

  // 2) A = x @ W1   (WMMA fp32)
  gemm16_wmma_kernel<NFEAT><<<cdiv(tiles, T / 32), T, 0, stream>>>(x, W1, A, tiles);
  if (tail)
    gemm16_tail_kernel<NFEAT><<<cdiv(tail * HID, T), T, 0, stream>>>(x, W1, A, tiles * 16, N);

  // 3) layer-1 aggregation into B, bias + relu
  self_loop_init_kernel<<<cdiv(N * HID, T), T, 0, stream>>>(A, dinv, B, N);
  edge_scatter_kernel<<<cdiv(E * HID, T), T, 0, stream>>>(ei, A, dinv, B, E);
  bias_relu_kernel<<<cdiv(N * HID, T), T, 0, stream>>>(B, b1, N);

  // 4) A = B @ W2   (WMMA fp32)
  gemm16_wmma_kernel<HID><<<cdiv(tiles, T / 32), T, 0, stream>>>(B, W2, A, tiles);
  if (tail)
    gemm16_tail_kernel<HID><<<cdiv(tail * HID, T), T, 0, stream>>>(B, W2, A, tiles * 16, N);

  // 5) layer-2 aggregation into B (h1 no longer needed), bias + relu
  self_loop_init_kernel<<<cdiv(N * HID, T), T, 0, stream>>>(A, dinv, B, N);
  edge_scatter_kernel<<<cdiv(E * HID, T), T, 0, stream>>>(ei, A, dinv, B, E);
  bias_relu_kernel<<<cdiv(N * HID, T), T, 0, stream>>>(B, b2, N);

  // 6) global add pool + MLP head
  pool_zero_kernel<<<cdiv(NGR * HID, T), T, 0, stream>>>(pooled);
  pool_kernel<<<cdiv(N * HID, T), T, 0, stream>>>(batch, B, pooled, N);
  mlp1_kernel<<<cdiv(NGR * MLPH, T), T, 0, stream>>>(pooled, lw1, lb1, hid);
  mlp2_kernel<<<cdiv(NGR * NCLS, T), T, 0, stream>>>(hid, lw2, lb2, out);

  (void)n_in; (void)out_size; (void)ws_size;
}